// QuantumDriverModel_62036507623827
// MI455X (gfx1250) — compile-verified
//
#include <hip/hip_runtime.h>
#include <hip/hip_bf16.h>
#include <math.h>

// ---------------- problem constants ----------------
#define N_OBS   16384
#define DIM     128
#define KCTX    16
#define QDIM    32
#define RANK    100
#define NDRV    4096
#define ELEMS   (DIM * DIM)          // 16384 per state
#define ALPHA   0.2f
#define EPS     1e-12f

typedef float v2f __attribute__((ext_vector_type(2)));
typedef float v8f __attribute__((ext_vector_type(8)));

// =====================================================================
// Kernel A: M_k = V_k V_k^T via V_WMMA_F32_16X16X4_F32
//   one wave (32 lanes) per 16x16 output tile; K-dim RANK=100 = 25*4
// A-matrix layout (16x4 f32): lanes 0-15 hold {K=0,K=1}, lanes 16-31 {K=2,K=3}
// B-matrix layout (4x16 f32): lane L<16 -> col L rows {K,K+1}; L>=16 -> col L-16 rows {K+2,K+3}
// C/D (16x16 f32): VGPR v, lane L -> row v + 8*(L>=16), col L%16
// =====================================================================
__global__ __launch_bounds__(32)
void profile_mm_kernel(const float* __restrict__ Vs, float* __restrict__ M) {
    const int it   = blockIdx.x;          // row tile (0..7)
    const int jt   = blockIdx.y;          // col tile (0..7)
    const int k    = blockIdx.z;          // profile index (0..15)
    const int lane = threadIdx.x;         // 0..31
    const int half = lane >> 4;
    const int l16  = lane & 15;
    const float* V = Vs + (size_t)k * DIM * RANK;
    const int i0 = it * 16, j0 = jt * 16;

    v8f acc = {0.f, 0.f, 0.f, 0.f, 0.f, 0.f, 0.f, 0.f};
    for (int r = 0; r < RANK; r += 4) {
        v2f a, b;
        // A[m][kk] = V[i0+m][r+kk]
        a.x = V[(i0 + l16) * RANK + r + 2 * half + 0];
        a.y = V[(i0 + l16) * RANK + r + 2 * half + 1];
        // B[kk][n] = V^T[r+kk][j0+n] = V[j0+n][r+kk]
        b.x = V[(j0 + l16) * RANK + r + 2 * half + 0];
        b.y = V[(j0 + l16) * RANK + r + 2 * half + 1];
        acc = __builtin_amdgcn_wmma_f32_16x16x4_f32(
            /*neg_a=*/false, a, /*neg_b=*/false, b,
            /*c_mod=*/(short)0, acc, /*reuse_a=*/false, /*reuse_b=*/false);
    }
    float* Mk = M + (size_t)k * ELEMS;
#pragma unroll
    for (int v = 0; v < 8; ++v) {
        const int row = i0 + v + half * 8;
        const int col = j0 + l16;
        Mk[row * DIM + col] = acc[v];
    }
}

// ------ trace of each M_k (16 blocks x 128 threads) ------
__global__ __launch_bounds__(128)
void trace_kernel(const float* __restrict__ M, float* __restrict__ tr) {
    const int k = blockIdx.x, t = threadIdx.x;
    __shared__ float red[128];
    red[t] = M[(size_t)k * ELEMS + t * (DIM + 1)];
    __syncthreads();
    for (int off = 64; off > 0; off >>= 1) {
        if (t < off) red[t] += red[t + off];
        __syncthreads();
    }
    if (t == 0) tr[k] = red[0];
}

// ------ profiles = M/max(tr,eps) (or I/D if tr<eps), in place ------
__global__ __launch_bounds__(256)
void normalize_kernel(float* __restrict__ M, const float* __restrict__ tr) {
    const int idx = blockIdx.x * 256 + threadIdx.x;   // 16*16384 total
    const int k = idx >> 14;
    const int e = idx & (ELEMS - 1);
    const float trk = tr[k];
    float val;
    if (trk < EPS)
        val = ((e >> 7) == (e & 127)) ? (1.0f / (float)DIM) : 0.0f;
    else
        val = M[idx] / fmaxf(trk, EPS);
    M[idx] = val;
}

// =====================================================================
// Kernel D: logits = C @ beta^T  ([16384,32]x[32,16]) via WMMA f32 16x16x4
// one wave per 16-row tile of C; the 16 output cols cover all K contexts
// =====================================================================
__global__ __launch_bounds__(32)
void logits_wmma_kernel(const float* __restrict__ C, const float* __restrict__ beta,
                        float* __restrict__ logits) {
    const int nt   = blockIdx.x;          // 0..1023
    const int lane = threadIdx.x;
    const int half = lane >> 4;
    const int l16  = lane & 15;
    const int n0   = nt * 16;

    v8f acc = {0.f, 0.f, 0.f, 0.f, 0.f, 0.f, 0.f, 0.f};
    for (int q = 0; q < QDIM; q += 4) {
        v2f a, b;
        a.x = C[(size_t)(n0 + l16) * QDIM + q + 2 * half + 0];
        a.y = C[(size_t)(n0 + l16) * QDIM + q + 2 * half + 1];
        // B[qq][kk] = beta[kk][qq]
        b.x = beta[l16 * QDIM + q + 2 * half + 0];
        b.y = beta[l16 * QDIM + q + 2 * half + 1];
        acc = __builtin_amdgcn_wmma_f32_16x16x4_f32(
            false, a, false, b, (short)0, acc, false, false);
    }
#pragma unroll
    for (int v = 0; v < 8; ++v) {
        const int row = n0 + v + half * 8;
        logits[row * KCTX + l16] = acc[v];
    }
}

// ------ rowwise softmax over K=16, in place ------
__global__ __launch_bounds__(256)
void softmax_kernel(float* __restrict__ pis) {
    const int n = blockIdx.x * 256 + threadIdx.x;     // < 16384
    float v[KCTX];
    float mx = -INFINITY;
#pragma unroll
    for (int k = 0; k < KCTX; ++k) { v[k] = pis[n * KCTX + k]; mx = fmaxf(mx, v[k]); }
    float s = 0.f;
#pragma unroll
    for (int k = 0; k < KCTX; ++k) { v[k] = expf(v[k] - mx); s += v[k]; }
    const float inv = 1.0f / s;
#pragma unroll
    for (int k = 0; k < KCTX; ++k) pis[n * KCTX + k] = v[k] * inv;
}

// =====================================================================
// Deterministic grouping of observations by driver (stable counting sort)
// =====================================================================
__global__ __launch_bounds__(256)
void count_kernel(const int* __restrict__ ids, int* __restrict__ counts) {
    const int d = blockIdx.x * 256 + threadIdx.x;     // < 4096
    int c = 0;
    for (int n = 0; n < N_OBS; ++n) c += (ids[n] == d) ? 1 : 0;
    counts[d] = c;
}

__global__ void prefix_kernel(const int* __restrict__ counts, int* __restrict__ offsets) {
    if (blockIdx.x == 0 && threadIdx.x == 0) {
        int acc = 0;
        for (int d = 0; d < NDRV; ++d) { offsets[d] = acc; acc += counts[d]; }
        offsets[NDRV] = acc;
    }
}

__global__ __launch_bounds__(256)
void fill_kernel(const int* __restrict__ ids, const int* __restrict__ offsets,
                 int* __restrict__ list) {
    const int d = blockIdx.x * 256 + threadIdx.x;     // < 4096
    int p = offsets[d];
    for (int n = 0; n < N_OBS; ++n)
        if (ids[n] == d) list[p++] = n;               // chronological within driver
}

// =====================================================================
// Kernel F: per-driver sequential density-matrix scan.
// One 256-thread block per driver. Each thread owns a CONTIGUOUS 64-elem
// chunk of the 128x128 state (half a row), held as 16x float4 in VGPRs:
//   - all state / profile traffic is 128-bit (b128) loads & stores
//   - phi_i is a per-thread constant (phiL[t>>1]); phi_c is a contiguous
//     64-entry window, so the Born readout factors phi_i out of the loop
// =====================================================================
__global__ __launch_bounds__(256)
void driver_scan_kernel(const float* __restrict__ states_in,
                        const float* __restrict__ Phi,
                        const float* __restrict__ pis,
                        const float* __restrict__ profiles,
                        const int* __restrict__ offsets,
                        const int* __restrict__ list,
                        const float* __restrict__ logit_eta_p,
                        float* __restrict__ states_out,
                        float* __restrict__ nll_per_driver) {
    const int d = blockIdx.x;
    const int t = threadIdx.x;
    __shared__ float phiL[DIM];
    __shared__ float piL[KCTX];
    __shared__ float red[256];

    const float eta = 1.0f / (1.0f + expf(-logit_eta_p[0]));
    const size_t base = (size_t)d * ELEMS + (size_t)t * 64;  // 256B-aligned

    float4 s[16];
    const float4* sin = (const float4*)(states_in + base);
#pragma unroll
    for (int j = 0; j < 16; ++j) s[j] = sin[j];

    const int row   = t >> 1;         // rho row handled by this thread
    const int cbase = (t & 1) * 64;   // rho column window base
    const float* __restrict__ profT = profiles + t * 64;

    const int beg = offsets[d];
    const int end = offsets[d + 1];
    float nll = 0.f;

    for (int o = beg; o < end; ++o) {
        const int n = list[o];
        if (t < DIM)  phiL[t] = Phi[(size_t)n * DIM + t];
        if (t < KCTX) piL[t]  = pis[(size_t)n * KCTX + t];
        __syncthreads();

        const float phi_i = phiL[row];
        float partial = 0.f;
#pragma unroll
        for (int j = 0; j < 16; ++j) {
            // mixture for this thread's elements [4j, 4j+4)
            float4 mix = {0.f, 0.f, 0.f, 0.f};
#pragma unroll
            for (int k = 0; k < KCTX; ++k) {
                const float4 pv = *(const float4*)(profT + (size_t)k * ELEMS + j * 4);
                const float pk = piL[k];
                mix.x = fmaf(pk, pv.x, mix.x);
                mix.y = fmaf(pk, pv.y, mix.y);
                mix.z = fmaf(pk, pv.z, mix.z);
                mix.w = fmaf(pk, pv.w, mix.w);
            }
            float4 rho;
            rho.x = fmaf(ALPHA, mix.x, (1.0f - ALPHA) * s[j].x);
            rho.y = fmaf(ALPHA, mix.y, (1.0f - ALPHA) * s[j].y);
            rho.z = fmaf(ALPHA, mix.z, (1.0f - ALPHA) * s[j].z);
            rho.w = fmaf(ALPHA, mix.w, (1.0f - ALPHA) * s[j].w);

            const float c0 = phiL[cbase + 4 * j + 0];
            const float c1 = phiL[cbase + 4 * j + 1];
            const float c2 = phiL[cbase + 4 * j + 2];
            const float c3 = phiL[cbase + 4 * j + 3];

            // Born readout partial (phi_i factored out, applied after loop)
            partial = fmaf(rho.x, c0, partial);
            partial = fmaf(rho.y, c1, partial);
            partial = fmaf(rho.z, c2, partial);
            partial = fmaf(rho.w, c3, partial);

            // rho_new = (1-eta)*rho + eta*phi_i*phi_c
            s[j].x = fmaf(eta, phi_i * c0, (1.0f - eta) * rho.x);
            s[j].y = fmaf(eta, phi_i * c1, (1.0f - eta) * rho.y);
            s[j].z = fmaf(eta, phi_i * c2, (1.0f - eta) * rho.z);
            s[j].w = fmaf(eta, phi_i * c3, (1.0f - eta) * rho.w);
        }
        red[t] = partial * phi_i;
        __syncthreads();
        for (int off = 128; off > 0; off >>= 1) {                // fixed-order tree
            if (t < off) red[t] += red[t + off];
            __syncthreads();
        }
        if (t == 0) nll += -logf(fmaxf(red[0], EPS));
        __syncthreads();                                          // before LDS reuse
    }

    if (t == 0) nll_per_driver[d] = nll;
    float4* sout = (float4*)(states_out + base);
#pragma unroll
    for (int j = 0; j < 16; ++j) sout[j] = s[j];
}

// ------ deterministic fixed-order reduction of 4096 per-driver NLLs ------
__global__ __launch_bounds__(256)
void nll_reduce_kernel(const float* __restrict__ nll_per_driver, float* __restrict__ out0) {
    __shared__ float red[256];
    const int t = threadIdx.x;
    float sum = 0.f;
    for (int i = 0; i < 16; ++i) sum += nll_per_driver[t * 16 + i];
    red[t] = sum;
    __syncthreads();
    for (int off = 128; off > 0; off >>= 1) {
        if (t < off) red[t] += red[t + off];
        __syncthreads();
    }
    if (t == 0) out0[0] = red[0];
}

// =====================================================================
// Host-side launch. Workspace layout (bytes):
//   profiles : [0, 4MB)                  16*128*128 f32 (M, then normalized in place)
//   pis      : [4MB, 5MB)                16384*16 f32 (logits, then softmax in place)
//   tr       : 5MB + 0                   16 f32 (padded to 256B)
//   counts   : 5MB + 256                 4096 i32
//   offsets  : +16KB                     4097 i32 (padded)
//   list     : +16.5KB                   16384 i32
//   nll      : +64KB                     4096 f32
// total ~5.2 MB
// =====================================================================
extern "C" void kernel_launch(void* const* d_in, const int* in_sizes, int n_in,
                              void* d_out, int out_size, void* d_ws, size_t ws_size,
                              hipStream_t stream) {
    const float* Phi    = (const float*)d_in[0];   // [N,128]
    const float* C      = (const float*)d_in[1];   // [N,32]
    const int*   ids    = (const int*)  d_in[2];   // [N]
    const float* st_in  = (const float*)d_in[3];   // [4096,128,128]
    const float* Vs     = (const float*)d_in[4];   // [16,128,100]
    const float* beta   = (const float*)d_in[5];   // [16,32]
    const float* logeta = (const float*)d_in[6];   // [1]

    char* ws = (char*)d_ws;
    float* profiles = (float*)(ws);                               // 4 MB
    float* pis      = (float*)(ws + (size_t)4 * 1024 * 1024);     // 1 MB
    char*  tail     = ws + (size_t)5 * 1024 * 1024;
    float* tr       = (float*)(tail);                             // 256 B slot
    int*   counts   = (int*)  (tail + 256);                       // 16 KB
    int*   offsets  = (int*)  (tail + 256 + 16384);               // ~16 KB
    int*   list     = (int*)  (tail + 256 + 16384 + 16640);       // 64 KB
    float* nlls     = (float*)(tail + 256 + 16384 + 16640 + 65536); // 16 KB

    float* out      = (float*)d_out;   // out[0] = sum nll ; out[1..] = states

    // 1. profiles: M_k = V_k V_k^T   (WMMA f32 16x16x4)
    profile_mm_kernel<<<dim3(8, 8, 16), 32, 0, stream>>>(Vs, profiles);
    trace_kernel<<<16, 128, 0, stream>>>(profiles, tr);
    normalize_kernel<<<(16 * ELEMS) / 256, 256, 0, stream>>>(profiles, tr);

    // 2. pis = softmax(C @ beta^T)   (WMMA f32 16x16x4 + rowwise softmax)
    logits_wmma_kernel<<<N_OBS / 16, 32, 0, stream>>>(C, beta, pis);
    softmax_kernel<<<N_OBS / 256, 256, 0, stream>>>(pis);

    // 3. stable grouping of observations by driver
    count_kernel<<<NDRV / 256, 256, 0, stream>>>(ids, counts);
    prefix_kernel<<<1, 1, 0, stream>>>(counts, offsets);
    fill_kernel<<<NDRV / 256, 256, 0, stream>>>(ids, offsets, list);

    // 4. per-driver sequential scan (one block per driver)
    driver_scan_kernel<<<NDRV, 256, 0, stream>>>(st_in, Phi, pis, profiles,
                                                 offsets, list, logeta,
                                                 out + 1, nlls);

    // 5. deterministic NLL sum
    nll_reduce_kernel<<<1, 256, 0, stream>>>(nlls, out);
}